// ParallelTransport_12661563589243
// MI455X (gfx1250) — compile-verified
//
#include <hip/hip_runtime.h>

// Problem constants (match reference)
#define B_     4
#define S_     512
#define D_     512
#define NN     64      // K*K = 8*8
#define NTERMS 6

typedef __attribute__((ext_vector_type(2))) float v2f;
typedef __attribute__((ext_vector_type(4))) float v4f;
typedef __attribute__((ext_vector_type(8))) float v8f;

// LDS layout (dynamic shared):
//  sA  : A[b,j] K-pair interleaved: elem (d,n) at (d>>1)*APAD2 + 2n + (d&1).
//        APAD2 = 96 (== 32 mod 64) -> half-wave b64 reads hit disjoint bank sets.
//  sEp : per-row M matrices, 512 rows x 64, row stride EPAD=66 (conflict-free)
//  sLj : 64 floats, lie_j[b,j,:,:]
#define APAD2 96
#define EPAD  66
#define SA_FLOATS  ((D_ / 2) * APAD2)   // 24576 (96 KB)
#define EP_FLOATS  (S_ * EPAD)          // 33792 (132 KB)
#define LJ_OFF     (SA_FLOATS + EP_FLOATS)
#define SMEM_FLOATS (LJ_OFF + NN)       // 58432 floats = 233728 bytes

__global__ __launch_bounds__(256, 1)
void ParallelTransport_12661563589243_kernel(const float* __restrict__ x,
                                             const float* __restrict__ A,
                                             float* __restrict__ out) {
  extern __shared__ float smem[];
  float* sA  = smem;
  float* sEp = smem + SA_FLOATS;
  float* sLj = smem + LJ_OFF;

  const int j    = blockIdx.x;
  const int b    = blockIdx.y;
  const int tid  = threadIdx.x;
  const int lane = tid & 31;
  const int wave = tid >> 5;
  const int g    = lane >> 4;   // half-wave group (lanes 0-15 vs 16-31)
  const int n16  = lane & 15;

  const float* xb  = x + (size_t)b * S_ * D_;
  const float* Abj = A + (((size_t)b * S_ + j) * (size_t)D_) * NN;

  // ---- Stage A[b,j] (512x64 f32) into LDS, K-pair interleaved ----
  // Task (h, p): sA[h*96 + 4p .. +3] = {A[2h][2p], A[2h+1][2p], A[2h][2p+1], A[2h+1][2p+1]}
  #pragma unroll
  for (int it = 0; it < 32; ++it) {
    int linear = tid + it * 256;        // 0..8191
    int p = linear & 31;                // n-pair index
    int h = linear >> 5;                // d-pair index, 0..255
    const float* s0 = Abj + (size_t)(2 * h) * NN + 2 * p;
    v2f e = *(const v2f*)s0;            // row 2h
    v2f o = *(const v2f*)(s0 + NN);     // row 2h+1
    v4f w;
    w.x = e.x; w.y = o.x; w.z = e.y; w.w = o.y;
    *(v4f*)(sA + h * APAD2 + 4 * p) = w;
  }
  __syncthreads();

  // ---- lie_j[b,j,kl] = sum_d x[b,j,d] * A2[d][kl]  (threads 0..63) ----
  if (tid < NN) {
    const float* xj = xb + (size_t)j * D_;
    float acc = 0.f;
    for (int d = 0; d < D_; ++d)
      acc = __builtin_fmaf(xj[d], sA[(d >> 1) * APAD2 + 2 * tid + (d & 1)], acc);
    sLj[tid] = acc;
  }
  // sLj consumers sync via the post-GEMM barrier.

  // ---- Single pass: wave owns 64 rows (4 row-tiles) x 64 cols (4 n-tiles) ----
  const int i0 = wave * 64;
  v8f acc[4][4] = {};   // [row-tile][n-tile], each 16x16 f32 (128 VGPRs)

  // A-fragment (X rows): lane holds M = n16, K = k0 + 2g + {0,1} -> float2
  const float* xr0 = xb + (size_t)(i0 + n16) * D_ + 2 * g;

  #pragma unroll 2
  for (int k0 = 0; k0 < D_; k0 += 4) {
    v2f a[4];
    #pragma unroll
    for (int rt = 0; rt < 4; ++rt)
      a[rt] = *(const v2f*)(xr0 + (size_t)(rt * 16) * D_ + k0);
    // B-fragment: lane holds N = n16, K = k0 + 2g + {0,1}; one b64 per n-tile
    const float* bp = sA + ((k0 >> 1) + g) * APAD2 + 2 * n16;
    v2f bf[4];
    #pragma unroll
    for (int nt = 0; nt < 4; ++nt)
      bf[nt] = *(const v2f*)(bp + nt * 32);
    #pragma unroll
    for (int nt = 0; nt < 4; ++nt)
      #pragma unroll
      for (int rt = 0; rt < 4; ++rt)
        acc[rt][nt] = __builtin_amdgcn_wmma_f32_16x16x4_f32(
            false, a[rt], false, bf[nt], (short)0, acc[rt][nt], false, false);
  }

  __syncthreads();   // Lj ready; GEMM done in all waves

  // ---- Scatter M = lie_j - lie_i into sEp (row stride EPAD) ----
  // C/D layout: VGPR r, lane -> (M = r + 8g, N = n16)
  #pragma unroll
  for (int rt = 0; rt < 4; ++rt)
    #pragma unroll
    for (int nt = 0; nt < 4; ++nt) {
      const float lj = sLj[nt * 16 + n16];
      #pragma unroll
      for (int r = 0; r < 8; ++r) {
        const int rowL = i0 + rt * 16 + r + 8 * g;
        sEp[rowL * EPAD + nt * 16 + n16] = lj - acc[rt][nt][r];
      }
    }
  __syncthreads();

  // ---- Taylor via Horner: T = I + (M/n) * T, n = 6..1. Rows tid and tid+256. ----
  #pragma unroll 1
  for (int rep = 0; rep < 2; ++rep) {
    const int row = rep * 256 + tid;
    const float* Mr = sEp + row * EPAD;   // 8x8 row-major, M[k][m] at k*8+m
    float T[64];
    #pragma unroll
    for (int e = 0; e < 64; ++e) T[e] = (e % 9 == 0) ? 1.f : 0.f;

    for (int n = NTERMS; n >= 1; --n) {
      const float inv = 1.0f / (float)n;
      float U[64];
      #pragma unroll
      for (int k = 0; k < 8; ++k) {
        #pragma unroll
        for (int l = 0; l < 8; ++l) U[k * 8 + l] = 0.f;
        #pragma unroll
        for (int m = 0; m < 8; ++m) {
          const float mk = Mr[k * 8 + m];
          #pragma unroll
          for (int l = 0; l < 8; ++l)
            U[k * 8 + l] = __builtin_fmaf(mk, T[m * 8 + l], U[k * 8 + l]);
        }
      }
      #pragma unroll
      for (int e = 0; e < 64; ++e)
        T[e] = ((e % 9 == 0) ? 1.f : 0.f) + U[e] * inv;
    }

    // ---- Write result row: out[b, i, j, 0:64], 16x float4 ----
    float* orow = out + (((size_t)b * S_ + row) * S_ + j) * NN;
    #pragma unroll
    for (int q = 0; q < 16; ++q) {
      v4f o;
      o.x = T[q * 4 + 0];
      o.y = T[q * 4 + 1];
      o.z = T[q * 4 + 2];
      o.w = T[q * 4 + 3];
      *(v4f*)(orow + q * 4) = o;
    }
  }
}

extern "C" void kernel_launch(void* const* d_in, const int* in_sizes, int n_in,
                              void* d_out, int out_size, void* d_ws, size_t ws_size,
                              hipStream_t stream) {
  (void)in_sizes; (void)n_in; (void)out_size; (void)d_ws; (void)ws_size;
  const float* x = (const float*)d_in[0];
  const float* A = (const float*)d_in[1];
  float* out = (float*)d_out;

  dim3 grid(S_, B_);                 // (j, b) = (512, 4) workgroups
  dim3 block(256);                   // 8 waves (wave32)
  size_t smem = SMEM_FLOATS * sizeof(float);  // ~228 KB of the 320 KB WGP LDS
  ParallelTransport_12661563589243_kernel<<<grid, block, smem, stream>>>(x, A, out);
}